// Calib_48180943127204
// MI455X (gfx1250) — compile-verified
//
#include <hip/hip_runtime.h>
#include <stdint.h>

#define NUM_CLASSES 19
#define TPB 256
#define HW_ (512 * 1024)
#define BATCH 2
#define NPIX (BATCH * HW_)          // 1,048,576 pixels
#define NBLK (NPIX / TPB)           // 4096 blocks
#define BLOCKS_PER_IMG (HW_ / TPB)  // 2048
#define IGNORE_LAB 255
#define THRESH 0.9f

// ---------------- CDNA5 async global->LDS copy (ASYNCcnt path) ----------------
// saddr form: global_load_async_to_lds_b32 vDstLdsByteOff, vByteOff, s[base:base+1]
static __device__ __forceinline__ void async_f32_to_lds(uint32_t lds_byte_off,
                                                        uint32_t v_byte_off,
                                                        const float* sbase) {
  asm volatile("global_load_async_to_lds_b32 %0, %1, %2 offset:0"
               :
               : "v"(lds_byte_off), "v"(v_byte_off), "s"(sbase)
               : "memory");
}

static __device__ __forceinline__ void wait_async0() {
  asm volatile("s_wait_asynccnt 0" ::: "memory");
}

// ---------------- kernels ----------------
__global__ void calib_init(unsigned* __restrict__ g_cnt) {
  if (threadIdx.x < NUM_CLASSES) g_cnt[threadIdx.x] = 0u;
}

__global__ __launch_bounds__(TPB) void calib_main(
    const float* __restrict__ pred,   // [B, C, H, W]
    const float* __restrict__ wgt,    // [C]
    const int* __restrict__ lab,      // [B, H, W]
    float* __restrict__ wp_out,       // [B, C, H, W]  (= d_out + 1)
    unsigned* __restrict__ g_cnt,     // [C]  integer atomics
    float* __restrict__ g_se)         // [NBLK] per-block partial sums
{
  __shared__ float tile[NUM_CLASSES * TPB];  // 19 KB staging tile
  __shared__ unsigned shCnt[NUM_CLASSES];
  __shared__ float shSe[TPB / 32];

  const int tid = threadIdx.x;
  // uniform decomposition: b and hw0 depend only on blockIdx.x
  const int b   = (int)(blockIdx.x / BLOCKS_PER_IMG);
  const int hw0 = (int)((blockIdx.x % BLOCKS_PER_IMG) * TPB);
  const int hw  = hw0 + tid;

  if (tid < NUM_CLASSES) shCnt[tid] = 0u;
  __syncthreads();

  // ---- issue 19 async channel loads into LDS (128B per wave per channel) ----
  const uint32_t ldsBase = (uint32_t)(uintptr_t)(&tile[0]);
  const uint32_t voff    = (uint32_t)hw * 4u;
#pragma unroll
  for (int c = 0; c < NUM_CLASSES; ++c) {
    const float* basec = pred + ((size_t)b * NUM_CLASSES + c) * (size_t)HW_;
    async_f32_to_lds(ldsBase + (uint32_t)(c * TPB + tid) * 4u, voff, basec);
  }
  wait_async0();  // each wave reads only the LDS it wrote -> no barrier needed

  // ---- read tile once into registers, softmax + reweight + renormalize ----
  float x[NUM_CLASSES];
  float m = -INFINITY;
#pragma unroll
  for (int c = 0; c < NUM_CLASSES; ++c) {
    x[c] = tile[c * TPB + tid];
    m = fmaxf(m, x[c]);
  }

  float e[NUM_CLASSES];
  float s = 0.0f;
#pragma unroll
  for (int c = 0; c < NUM_CLASSES; ++c) {
    float v = __expf(x[c] - m) * wgt[c];
    e[c] = v;
    s += v;
  }
  const float inv = 1.0f / s;

  const int L = lab[b * HW_ + hw];
  float sumsq = 0.0f;
  float wl = 0.0f;
#pragma unroll
  for (int c = 0; c < NUM_CLASSES; ++c) {
    float w = e[c] * inv;
    // streaming 80MB output: nontemporal store
    __builtin_nontemporal_store(w, &wp_out[((size_t)b * NUM_CLASSES + c) * (size_t)HW_ + hw]);
    sumsq += w * w;
    if (w > THRESH) atomicAdd(&shCnt[c], 1u);  // at most one class/pixel; LDS atomic
    if (c == L) wl = w;
  }

  float se;
  if (L == IGNORE_LAB)                 se = 0.0f;
  else if (L >= 0 && L < NUM_CLASSES)  se = sumsq - 2.0f * wl + 1.0f;  // ||wp - onehot||^2
  else                                 se = sumsq;                      // oob label -> zero onehot

  // deterministic per-block reduction: wave shuffle, then fixed-order combine
#pragma unroll
  for (int off = 16; off > 0; off >>= 1) se += __shfl_down(se, off, 32);
  if ((tid & 31) == 0) shSe[tid >> 5] = se;
  __syncthreads();

  if (tid == 0) {
    float t = 0.0f;
#pragma unroll
    for (int w = 0; w < TPB / 32; ++w) t += shSe[w];
    g_se[blockIdx.x] = t;
  }
  if (tid < NUM_CLASSES) {
    unsigned c = shCnt[tid];
    if (c) atomicAdd(&g_cnt[tid], c);  // integer: exact & deterministic
  }
}

__global__ __launch_bounds__(TPB) void calib_final(
    const unsigned* __restrict__ g_cnt,
    const float* __restrict__ g_se,
    float* __restrict__ out)  // out[0] = all_loss
{
  __shared__ float shSe[TPB / 32];
  const int tid = threadIdx.x;

  // fixed-order strided accumulation -> deterministic
  float s = 0.0f;
  for (int i = tid; i < NBLK; i += TPB) s += g_se[i];
#pragma unroll
  for (int off = 16; off > 0; off >>= 1) s += __shfl_down(s, off, 32);
  if ((tid & 31) == 0) shSe[tid >> 5] = s;
  __syncthreads();

  if (tid == 0) {
    float tot_se = 0.0f;
#pragma unroll
    for (int w = 0; w < TPB / 32; ++w) tot_se += shSe[w];

    const float prior[NUM_CLASSES] = {
        32.07f, 5.71f, 20.7f, 0.564f, 0.761f, 1.054f, 0.1696f, 0.5014f,
        13.4993f, 0.8981f, 3.6445f, 1.1458f, 0.1393f, 6.0f, 0.2949f,
        0.1954f, 0.2341f, 0.0818f, 0.3917f};
    float ps = 0.0f;
    for (int c = 0; c < NUM_CLASSES; ++c) ps += prior[c];

    float tot = 0.0f;
    float cnt[NUM_CLASSES];
    for (int c = 0; c < NUM_CLASSES; ++c) {
      cnt[c] = (float)g_cnt[c];
      tot += cnt[c];
    }
    float rl = 0.0f;
    for (int c = 0; c < NUM_CLASSES; ++c) {
      float d = cnt[c] / tot - prior[c] / ps;
      rl += d * d;
    }
    const float mse = tot_se / (float)NPIX;
    out[0] = rl + 0.05f * mse;
  }
}

// ---------------- entry ----------------
extern "C" void kernel_launch(void* const* d_in, const int* in_sizes, int n_in,
                              void* d_out, int out_size, void* d_ws, size_t ws_size,
                              hipStream_t stream) {
  (void)in_sizes; (void)n_in; (void)out_size; (void)ws_size;
  const float* pred = (const float*)d_in[0];
  const float* wgt  = (const float*)d_in[1];
  const int*   lab  = (const int*)d_in[2];
  float* out = (float*)d_out;  // out[0] = loss, out+1 = wp (flat, tuple order)

  unsigned* g_cnt = (unsigned*)d_ws;                       // 19 * 4 B
  float*    g_se  = (float*)((char*)d_ws + 256);           // NBLK * 4 B = 16 KB

  calib_init<<<1, 32, 0, stream>>>(g_cnt);
  calib_main<<<NBLK, TPB, 0, stream>>>(pred, wgt, lab, out + 1, g_cnt, g_se);
  calib_final<<<1, TPB, 0, stream>>>(g_cnt, g_se, out);
}